// Generator_59261958750907
// MI455X (gfx1250) — compile-verified
//
#include <hip/hip_runtime.h>
#include <math.h>

#define N_BATCH 1024
#define L_SIM   4096
#define P_ACQ   8192

typedef __attribute__((ext_vector_type(2))) float v2f;
typedef __attribute__((ext_vector_type(8))) float v8f;

static __device__ __forceinline__ v8f wmma4(v2f a, v2f b, v8f c) {
  // D(16x16,f32) = A(16x4,f32) x B(4x16,f32) + C
  return __builtin_amdgcn_wmma_f32_16x16x4_f32(false, a, false, b, (short)0, c,
                                               false, false);
}

// ---------------------------------------------------------------------------
// Stage A: bounded random walk -> smooth -> normalize -> resample to ppm axis
// one block (256 threads) per sample; everything in LDS
// ---------------------------------------------------------------------------
__global__ __launch_bounds__(256)
void prep_kernel(const float* __restrict__ startv,
                 const float* __restrict__ endv,
                 const float* __restrict__ stdv,
                 const float* __restrict__ lowv,
                 const float* __restrict__ upv,
                 const int*   __restrict__ winv,
                 const float* __restrict__ scalev,
                 const float* __restrict__ noise,
                 const float* __restrict__ omitv,
                 const float* __restrict__ ppm,
                 const float* __restrict__ ppmr,
                 float* __restrict__ acq)
{
  __shared__ float r[L_SIM];
  __shared__ float cs[L_SIM + 1];
  __shared__ float pA[256];
  __shared__ float pB[256];

  const int n   = blockIdx.x;
  const int tid = threadIdx.x;
  const float sd  = stdv[n];
  const float st  = startv[n];
  const float en  = endv[n];
  const float lo  = lowv[n];
  const float up  = upv[n];
  const int   w   = winv[n];
  const float oms = omitv[n] * scalev[n];
  const float inv4095 = 1.0f / 4095.0f;

  float loc[16];
  const int base = tid * 16;

  // ---- rand = cumsum(std * (noise - 0.5)) ----
  {
    float s = 0.0f;
#pragma unroll
    for (int j = 0; j < 16; ++j) {
      float v = sd * (noise[(size_t)n * L_SIM + base + j] - 0.5f);
      s += v;
      loc[j] = s;                       // inclusive within chunk
    }
    pA[tid] = s;
    __syncthreads();
    for (int off = 1; off < 256; off <<= 1) {   // Hillis-Steele inclusive scan
      float add = (tid >= off) ? pA[tid - off] : 0.0f;
      __syncthreads();
      pA[tid] += add;
      __syncthreads();
    }
    float pre = pA[tid] - s;            // exclusive prefix
#pragma unroll
    for (int j = 0; j < 16; ++j) r[base + j] = pre + loc[j];
  }
  __syncthreads();

  const float r0 = r[0], rN = r[L_SIM - 1];

  // ---- min/max of detrended deltas ----
  {
    float mx = -3.4e38f, mn = 3.4e38f;
#pragma unroll
    for (int j = 0; j < 16; ++j) {
      int i = base + j;
      float d = r[i] - (r0 + (rN - r0) * (float)i * inv4095);
      mx = fmaxf(mx, d);
      mn = fminf(mn, d);
    }
    pA[tid] = mx;
    pB[tid] = mn;
    __syncthreads();
    for (int off = 128; off > 0; off >>= 1) {
      if (tid < off) {
        pA[tid] = fmaxf(pA[tid], pA[tid + off]);
        pB[tid] = fminf(pB[tid], pB[tid + off]);
      }
      __syncthreads();
    }
  }
  const float span    = pA[0] - pB[0];
  const float bound   = up - lo;
  const float squeeze = 1.0f / fmaxf(1.0f, span / bound);

  // ---- walk with reflecting bounds (in place) ----
#pragma unroll
  for (int j = 0; j < 16; ++j) {
    int i = base + j;
    float tr = st + (en - st) * (float)i * inv4095;
    float d  = (r[i] - (r0 + (rN - r0) * (float)i * inv4095)) * squeeze;
    float ub = up - tr, lb = lo - tr;
    float over = d - ub;
    d = (over >= 0.0f) ? (ub - over) : d;
    float under = lb - d;
    d = (under >= 0.0f) ? (lb + under) : d;
    r[i] = tr + d;
  }
  __syncthreads();

  // ---- cumsum of walk -> cs (cs[0]=0) ----
  {
    float s = 0.0f;
#pragma unroll
    for (int j = 0; j < 16; ++j) { s += r[base + j]; loc[j] = s; }
    pA[tid] = s;
    __syncthreads();
    for (int off = 1; off < 256; off <<= 1) {
      float add = (tid >= off) ? pA[tid - off] : 0.0f;
      __syncthreads();
      pA[tid] += add;
      __syncthreads();
    }
    float pre = pA[tid] - s;
#pragma unroll
    for (int j = 0; j < 16; ++j) cs[base + j + 1] = pre + loc[j];
    if (tid == 0) cs[0] = 0.0f;
  }
  __syncthreads();

  // ---- box smooth via clamped cumsum (zero padding == clamp) ----
  const float invw = 1.0f / (float)w;
#pragma unroll
  for (int j = 0; j < 16; ++j) {
    int i    = base + j;
    int lo_i = i - (w >> 1);
    int hi_i = lo_i + w;
    lo_i = min(max(lo_i, 0), L_SIM);
    hi_i = min(max(hi_i, 0), L_SIM);
    r[i] = (cs[hi_i] - cs[lo_i]) * invw;
  }
  __syncthreads();

  const float b0 = r[0], bN = r[L_SIM - 1];

  // ---- max|b - trend| ----
  {
    float mx = 0.0f;
#pragma unroll
    for (int j = 0; j < 16; ++j) {
      int i = base + j;
      float e = r[i] - (b0 + (bN - b0) * (float)i * inv4095);
      mx = fmaxf(mx, fabsf(e));
    }
    pA[tid] = mx;
    __syncthreads();
    for (int off = 128; off > 0; off >>= 1) {
      if (tid < off) pA[tid] = fmaxf(pA[tid], pA[tid + off]);
      __syncthreads();
    }
  }
  const float den = pA[0];
  const float nrm = (den == 0.0f) ? 1.0f : (1.0f / den);
  __syncthreads();

  // ---- detrend, normalize, omit, scale (in place) ----
#pragma unroll
  for (int j = 0; j < 16; ++j) {
    int i = base + j;
    float e = r[i] - (b0 + (bN - b0) * (float)i * inv4095);
    r[i] = e * nrm * oms;
  }
  __syncthreads();

  // ---- linear resample onto acquired ppm axis (zero outside range) ----
  const float plo = ppmr[0], phi = ppmr[1];
  const float toIdx = 4095.0f / (phi - plo);
  for (int p = tid; p < P_ACQ; p += 256) {
    float x = ppm[p];
    float v = 0.0f;
    if (x >= plo && x <= phi) {
      float u = (x - plo) * toIdx;
      int i0 = (int)floorf(u);
      i0 = min(max(i0, 0), L_SIM - 1);
      int i1 = min(i0 + 1, L_SIM - 1);
      float f = u - (float)i0;
      v = r[i0] * (1.0f - f) + r[i1] * f;
    }
    acq[(size_t)n * P_ACQ + p] = v;
  }
}

// ---------------------------------------------------------------------------
// WMMA four-step FFT stages (8192 = 64 x 128), data resident in LDS.
// inner: Y[n1][k2] = sum_n2 x[n1 + 64*n2] * W128[n2*k2]   (A=data, B=twiddle)
// outer: X[k1*128+k2] = sum_n1 W64[k1*n1] * Z[n1][k2]      (A=twiddle, B=data)
// Twiddle tables hold the FORWARD sign; simag=+1 fwd, -1 inverse (conjugate).
// Each wave owns TWO column tiles of the SAME row so the row-shared A
// fragment is fetched from LDS once per K-step and feeds 4-8 WMMAs.
// ---------------------------------------------------------------------------
static __device__ void fft_inner(const float* __restrict__ xre,
                                 const float* __restrict__ xim,
                                 bool imZero,
                                 float* __restrict__ yre,
                                 float* __restrict__ yim,
                                 const float* __restrict__ w128re,
                                 const float* __restrict__ w128im,
                                 float simag)
{
  const int tid  = threadIdx.x;
  const int wave = tid >> 5;
  const int lane = tid & 31;
  const int half = lane >> 4;       // 0 or 1
  const int l15  = lane & 15;
  const int kh   = half << 1;       // K sub-offset 0 or 2 (f32 A/B layout)

  const int row = wave & 3;         // n1 tile row (4 rows of 16)
  const int c0  = wave >> 2;        // column-tile pair {c0, c0+4} of 8
  const int m0  = row << 4;
  const int n0a = c0 << 4;
  const int n0b = (c0 + 4) << 4;
  const int m   = m0 + l15;
  const int na  = n0a + l15;
  const int nb  = n0b + l15;

  v8f ar0 = {}, ai0 = {}, ar1 = {}, ai1 = {};

  for (int k0 = 0; k0 < 128; k0 += 4) {
    const int ka = k0 + kh;
    v2f a;                                    // shared A (data) fragment
    a.x = xre[m + 64 * ka];
    a.y = xre[m + 64 * (ka + 1)];

    const int ia0 = (ka * na) & 127;
    const int ia1 = ((ka + 1) * na) & 127;
    const int ib0 = (ka * nb) & 127;
    const int ib1 = ((ka + 1) * nb) & 127;
    v2f bra, bia, brb, bib;
    bra.x = w128re[ia0];         bra.y = w128re[ia1];
    bia.x = simag * w128im[ia0]; bia.y = simag * w128im[ia1];
    brb.x = w128re[ib0];         brb.y = w128re[ib1];
    bib.x = simag * w128im[ib0]; bib.y = simag * w128im[ib1];

    ar0 = wmma4(a, bra, ar0);
    ai0 = wmma4(a, bia, ai0);
    ar1 = wmma4(a, brb, ar1);
    ai1 = wmma4(a, bib, ai1);
    if (!imZero) {
      v2f a2;                                 // shared imag-plane fragment
      a2.x = xim[m + 64 * ka];
      a2.y = xim[m + 64 * (ka + 1)];
      v2f nbia, nbib;
      nbia.x = -bia.x; nbia.y = -bia.y;
      nbib.x = -bib.x; nbib.y = -bib.y;
      ar0 = wmma4(a2, nbia, ar0);
      ai0 = wmma4(a2, bra, ai0);
      ar1 = wmma4(a2, nbib, ar1);
      ai1 = wmma4(a2, brb, ai1);
    }
  }
  const int rbase = m0 + (half << 3);         // D layout: lanes>=16 -> rows +8
#pragma unroll
  for (int rr = 0; rr < 8; ++rr) {
    const int rowi = (rbase + rr) * 128;
    yre[rowi + n0a + l15] = ar0[rr];
    yim[rowi + n0a + l15] = ai0[rr];
    yre[rowi + n0b + l15] = ar1[rr];
    yim[rowi + n0b + l15] = ai1[rr];
  }
}

static __device__ void fft_outer(const float* __restrict__ zre,
                                 const float* __restrict__ zim,
                                 float* __restrict__ Xre,
                                 float* __restrict__ Xim,
                                 const float* __restrict__ w64re,
                                 const float* __restrict__ w64im,
                                 float simag, float scl)
{
  const int tid  = threadIdx.x;
  const int wave = tid >> 5;
  const int lane = tid & 31;
  const int half = lane >> 4;
  const int l15  = lane & 15;
  const int kh   = half << 1;

  const int row = wave & 3;         // k1 tile row
  const int c0  = wave >> 2;        // column-tile pair {c0, c0+4}
  const int m0  = row << 4;
  const int n0a = c0 << 4;
  const int n0b = (c0 + 4) << 4;
  const int m   = m0 + l15;
  const int na  = n0a + l15;
  const int nb  = n0b + l15;

  v8f ar0 = {}, ai0 = {}, ar1 = {}, ai1 = {};

  for (int k0 = 0; k0 < 64; k0 += 4) {
    const int ka = k0 + kh;
    const int i0 = (m * ka) & 63;
    const int i1 = (m * (ka + 1)) & 63;
    v2f tr, ti, nti;                          // shared A (twiddle) fragment
    tr.x = w64re[i0];          tr.y = w64re[i1];
    ti.x = simag * w64im[i0];  ti.y = simag * w64im[i1];
    nti.x = -ti.x;             nti.y = -ti.y;

    v2f bra, bia, brb, bib;
    bra.x = zre[ka * 128 + na]; bra.y = zre[(ka + 1) * 128 + na];
    bia.x = zim[ka * 128 + na]; bia.y = zim[(ka + 1) * 128 + na];
    brb.x = zre[ka * 128 + nb]; brb.y = zre[(ka + 1) * 128 + nb];
    bib.x = zim[ka * 128 + nb]; bib.y = zim[(ka + 1) * 128 + nb];

    ar0 = wmma4(tr, bra, ar0);    // Re += Tre*Bre
    ar0 = wmma4(nti, bia, ar0);   // Re -= Tim*Bim
    ai0 = wmma4(tr, bia, ai0);    // Im += Tre*Bim
    ai0 = wmma4(ti, bra, ai0);    // Im += Tim*Bre
    ar1 = wmma4(tr, brb, ar1);
    ar1 = wmma4(nti, bib, ar1);
    ai1 = wmma4(tr, bib, ai1);
    ai1 = wmma4(ti, brb, ai1);
  }
  const int rbase = m0 + (half << 3);
#pragma unroll
  for (int rr = 0; rr < 8; ++rr) {
    const int rowi = (rbase + rr) * 128;
    Xre[rowi + n0a + l15] = scl * ar0[rr];
    Xim[rowi + n0a + l15] = scl * ai0[rr];
    Xre[rowi + n0b + l15] = scl * ar1[rr];
    Xim[rowi + n0b + l15] = scl * ai1[rr];
  }
}

// ---------------------------------------------------------------------------
// Stage B: FFT-based analytic signal per sample, fully in LDS.
// one block (512 threads = 16 waves) per sample.
// ---------------------------------------------------------------------------
__global__ __launch_bounds__(512)
void hilbert_kernel(const float* __restrict__ acq, float* __restrict__ out)
{
  extern __shared__ float smem[];
  float* Are    = smem;                 // 8192
  float* Aim    = Are + P_ACQ;          // 8192
  float* Tre    = Aim + P_ACQ;          // 8192
  float* Tim    = Tre + P_ACQ;          // 8192
  float* w128re = Tim + P_ACQ;          // 128
  float* w128im = w128re + 128;         // 128
  float* w64re  = w128im + 128;         // 64
  float* w64im  = w64re + 64;           // 64

  const int n   = blockIdx.x;
  const int tid = threadIdx.x;
  const float TWO_PI = 6.28318530717958647692f;

  if (tid < 128) {
    float s, c;
    __sincosf(-TWO_PI * (float)tid * (1.0f / 128.0f), &s, &c);
    w128re[tid] = c;
    w128im[tid] = s;
    if (tid < 64) {
      __sincosf(-TWO_PI * (float)tid * (1.0f / 64.0f), &s, &c);
      w64re[tid] = c;
      w64im[tid] = s;
    }
  }
  for (int i = tid; i < P_ACQ; i += 512) {
    Are[i] = acq[(size_t)n * P_ACQ + i];
    Aim[i] = 0.0f;
  }
  __syncthreads();

  // ---- forward FFT: A -> T -> (twiddle) -> A ----
  fft_inner(Are, Aim, true, Tre, Tim, w128re, w128im, 1.0f);
  __syncthreads();
  for (int i = tid; i < P_ACQ; i += 512) {
    int n1 = i >> 7, k2 = i & 127;
    float s, c;
    __sincosf(-TWO_PI * (float)(n1 * k2) * (1.0f / 8192.0f), &s, &c);
    float re = Tre[i], im = Tim[i];
    Tre[i] = re * c - im * s;
    Tim[i] = re * s + im * c;
  }
  __syncthreads();
  fft_outer(Tre, Tim, Are, Aim, w64re, w64im, 1.0f, 1.0f);
  __syncthreads();

  // ---- analytic-signal filter: h = [1, 2..., 1, 0...] ----
  for (int i = tid; i < P_ACQ; i += 512) {
    float h = (i == 0 || i == (P_ACQ / 2)) ? 1.0f
                                           : (i < (P_ACQ / 2) ? 2.0f : 0.0f);
    Are[i] *= h;
    Aim[i] *= h;
  }
  __syncthreads();

  // ---- inverse FFT (conjugated twiddles, 1/P scale) ----
  fft_inner(Are, Aim, false, Tre, Tim, w128re, w128im, -1.0f);
  __syncthreads();
  for (int i = tid; i < P_ACQ; i += 512) {
    int k1 = i >> 7, m2 = i & 127;
    float s, c;
    __sincosf(TWO_PI * (float)(k1 * m2) * (1.0f / 8192.0f), &s, &c);
    float re = Tre[i], im = Tim[i];
    Tre[i] = re * c - im * s;
    Tim[i] = re * s + im * c;
  }
  __syncthreads();
  fft_outer(Tre, Tim, Are, Aim, w64re, w64im, -1.0f, 1.0f / 8192.0f);
  __syncthreads();

  // ---- write (flip(raw), raw), each [N,2,P] ----
  const size_t NP2 = (size_t)N_BATCH * 2 * P_ACQ;
  float* flipd = out;
  float* raw   = out + NP2;
  const size_t base = (size_t)n * 2 * P_ACQ;
  for (int i = tid; i < P_ACQ; i += 512) {
    float re = Are[i], im = Aim[i];
    raw[base + i]                         = re;
    raw[base + P_ACQ + i]                 = im;
    flipd[base + (P_ACQ - 1 - i)]         = re;
    flipd[base + P_ACQ + (P_ACQ - 1 - i)] = im;
  }
}

// ---------------------------------------------------------------------------
extern "C" void kernel_launch(void* const* d_in, const int* in_sizes, int n_in,
                              void* d_out, int out_size, void* d_ws,
                              size_t ws_size, hipStream_t stream)
{
  (void)in_sizes; (void)n_in; (void)out_size; (void)ws_size;

  const float* startv = (const float*)d_in[0];
  const float* endv   = (const float*)d_in[1];
  const float* stdv   = (const float*)d_in[2];
  const float* lowv   = (const float*)d_in[3];
  const float* upv    = (const float*)d_in[4];
  const int*   winv   = (const int*)d_in[5];
  const float* scalev = (const float*)d_in[6];
  const float* noise  = (const float*)d_in[7];
  const float* omitv  = (const float*)d_in[8];
  const float* ppm    = (const float*)d_in[9];
  const float* ppmr   = (const float*)d_in[10];

  float* acq = (float*)d_ws;              // N*P floats staging buffer
  float* out = (float*)d_out;

  prep_kernel<<<N_BATCH, 256, 0, stream>>>(startv, endv, stdv, lowv, upv, winv,
                                           scalev, noise, omitv, ppm, ppmr, acq);

  const size_t shmem = (size_t)(4 * P_ACQ + 2 * 128 + 2 * 64) * sizeof(float);
  hilbert_kernel<<<N_BATCH, 512, shmem, stream>>>(acq, out);
}